// UnrolledSIR_40003325395182
// MI455X (gfx1250) — compile-verified
//
#include <hip/hip_runtime.h>

// SIR Euler integrator: 364 days x 100 substeps, strictly serial recurrence.
// Latency-bound: one wave, one active lane, 2-dependent-VALU-op chain per step.
// WMMA / TDM intentionally unused: no matrix work, no bulk data movement exists
// in this problem (4.4 KB total output, ~0.18 MFLOP, fully sequential).

#define SIR_T      365
#define SIR_STEPS  100

__global__ __launch_bounds__(32, 1)
void sir_euler_kernel(const float* __restrict__ y0,
                      const float* __restrict__ beta,
                      float* __restrict__ out)
{
    // Single lane carries the recurrence; other 31 lanes of the wave retire.
    if (threadIdx.x != 0) return;

    const float h  = 1.0f / (float)SIR_STEPS;  // 0.01f (exact f32 of reference)
    const float g  = 0.1f;
    const float b  = beta[0];                  // beta[-1] with len-1 array
    const float bh = b * h;                    // folded infection coefficient
    const float gh = g * h;                    // folded recovery coefficient

    float S = y0[0];
    float I = y0[1];
    float R = y0[2];

    for (int day = 0; day < SIR_T - 1; ++day) {
        #pragma unroll
        for (int s = 0; s < SIR_STEPS; ++s) {
            // s2i = h*b*I*S ; i2r = h*g*I
            // S' = S - s2i
            // I' = I + s2i - i2r = (1 - gh)*I + bh*t
            // R' = R + i2r
            float t  = S * I;                        // critical op 1
            float q  = __builtin_fmaf(-gh, I, I);    // off critical path
            float Rn = __builtin_fmaf( gh, I, R);    // off critical path
            float Sn = __builtin_fmaf(-bh, t, S);    // critical op 2
            float In = __builtin_fmaf( bh, t, q);    // critical op 2 (parallel)
            S = Sn; I = In; R = Rn;
        }
        // End-of-day state -> row `day` (reference drops the initial condition).
        // Three consecutive dwords from one lane -> single global_store_b96.
        out[day * 3 + 0] = S;
        out[day * 3 + 1] = I;
        out[day * 3 + 2] = R;
    }
}

extern "C" void kernel_launch(void* const* d_in, const int* in_sizes, int n_in,
                              void* d_out, int out_size, void* d_ws, size_t ws_size,
                              hipStream_t stream)
{
    (void)in_sizes; (void)n_in; (void)out_size; (void)d_ws; (void)ws_size;

    const float* y0   = (const float*)d_in[0];   // 3 floats
    const float* beta = (const float*)d_in[1];   // 1 float
    float*       out  = (float*)d_out;           // [364, 3] float32

    sir_euler_kernel<<<dim3(1), dim3(32), 0, stream>>>(y0, beta, out);
}